// Top1Router_79413945303480
// MI455X (gfx1250) — compile-verified
//
#include <hip/hip_runtime.h>
#include <math.h>

#define EXPERTS 64

typedef __attribute__((ext_vector_type(4))) float v4f;
typedef __attribute__((address_space(1))) int as1_int;
typedef __attribute__((address_space(3))) int as3_int;

// ---------------------------------------------------------------------------
// Kernel 1: one wave32 per token. Each lane holds 2 of the 64 logits.
// Cross-lane shfl_xor reduction for (max, argmax-first-occurrence) and sum-exp.
// gate = softmax[top1] = 1 / sum(exp(x - max)).
// ---------------------------------------------------------------------------
__global__ void router_gate_kernel(const float* __restrict__ in,
                                   int* __restrict__ top1,
                                   float* __restrict__ gate,
                                   int S) {
  int gtid  = blockIdx.x * blockDim.x + threadIdx.x;
  int token = gtid >> 5;
  int lane  = threadIdx.x & 31;
  if (token >= S) return;

  const float* row = in + (size_t)token * EXPERTS;
  float x0 = row[lane * 2 + 0];
  float x1 = row[lane * 2 + 1];

  // local max with first-occurrence tie-break (x0 index < x1 index)
  float m  = (x0 >= x1) ? x0 : x1;
  int  idx = (x0 >= x1) ? (lane * 2) : (lane * 2 + 1);

#pragma unroll
  for (int off = 16; off > 0; off >>= 1) {
    float om = __shfl_xor(m, off, 32);
    int   oi = __shfl_xor(idx, off, 32);
    if (om > m || (om == m && oi < idx)) { m = om; idx = oi; }
  }

  float se = expf(x0 - m) + expf(x1 - m);
#pragma unroll
  for (int off = 16; off > 0; off >>= 1) se += __shfl_xor(se, off, 32);

  if (lane == 0) {
    top1[token] = idx;
    gate[token] = 1.0f / se;  // exp(0) of the max element is included in se
  }
}

// ---------------------------------------------------------------------------
// Kernel 2: one block per expert. Ordered exclusive scan over the 8192-token
// sequence. Each 1KB chunk of top1[] is staged into LDS via the CDNA5 async
// global->LDS path (GLOBAL_LOAD_ASYNC_TO_LDS_B32, ASYNCcnt-tracked), then the
// rank is computed with wave32 ballot+popc prefix inside each wave, wave
// totals combined through LDS, running base carried across chunks. Each
// token's slot is written exactly once (by its top-1 expert's block): rank,
// or -1 if the expert is over capacity.
// ---------------------------------------------------------------------------
__global__ void rank_scan_kernel(const int* __restrict__ top1,
                                 int* __restrict__ slot,
                                 int S, int cap) {
  __shared__ int s_top1[256];  // staged chunk of top1[]
  __shared__ int s_wtot[8];    // blockDim.x == 256 -> 8 waves
  __shared__ int s_base;
  const int expert = blockIdx.x;
  const int lane   = threadIdx.x & 31;
  const int wave   = threadIdx.x >> 5;
  const int nwaves = blockDim.x >> 5;

  if (threadIdx.x == 0) s_base = 0;
  __syncthreads();

  for (int start = 0; start < S; start += blockDim.x) {
    int t = start + threadIdx.x;
    if (t < S) {
      // async copy: each lane copies global top1[t] into its LDS slot s_top1[threadIdx.x]  (ASYNCcnt-tracked)
      __builtin_amdgcn_global_load_async_to_lds_b32(
          (as1_int*)(top1 + t),
          (as3_int*)(&s_top1[threadIdx.x]),
          0, 0);
    }
    asm volatile("s_wait_asynccnt 0x0" ::: "memory");
    __syncthreads();

    int flag = (t < S && s_top1[threadIdx.x] == expert) ? 1 : 0;

    unsigned long long bal = __ballot(flag);  // wave32: low 32 bits valid
    int wprefix = __popcll(bal & ((1ull << lane) - 1ull));
    int wtot    = __popcll(bal);
    if (lane == 0) s_wtot[wave] = wtot;
    __syncthreads();

    int offset = s_base;
    for (int w = 0; w < wave; ++w) offset += s_wtot[w];
    if (flag) {
      int rank = offset + wprefix;
      slot[t] = (rank < cap) ? rank : -1;
    }
    __syncthreads();

    if (threadIdx.x == 0) {
      int tot = 0;
      for (int w = 0; w < nwaves; ++w) tot += s_wtot[w];
      s_base += tot;
    }
    __syncthreads();  // also protects s_top1 before next chunk's async writes
  }
}

// ---------------------------------------------------------------------------
// Kernel 3: zero-fill both outputs (671 MB) with non-temporal B128 stores.
// Output is 3.5x the 192MB L2 -> NT streaming stores avoid cache thrash.
// ---------------------------------------------------------------------------
__global__ void zero_fill_kernel(float* __restrict__ out, long long n4) {
  long long i      = (long long)blockIdx.x * blockDim.x + threadIdx.x;
  long long stride = (long long)gridDim.x * blockDim.x;
  v4f z = {0.f, 0.f, 0.f, 0.f};
  v4f* p = (v4f*)out;
  for (; i < n4; i += stride) {
    __builtin_nontemporal_store(z, p + i);
  }
}

// ---------------------------------------------------------------------------
// Kernel 4: scatter the single nonzero per kept token into both halves.
// Runs after the fill (stream ordering).
// ---------------------------------------------------------------------------
__global__ void scatter_kernel(const int* __restrict__ top1,
                               const float* __restrict__ gate,
                               const int* __restrict__ slot,
                               float* __restrict__ out,
                               int S, int cap, size_t combineElems) {
  int t = blockIdx.x * blockDim.x + threadIdx.x;
  if (t >= S) return;
  int sl = slot[t];
  if (sl < 0) return;
  size_t idx = ((size_t)t * EXPERTS + (size_t)top1[t]) * (size_t)cap + (size_t)sl;
  out[idx]                = gate[t];  // combine_weights
  out[combineElems + idx] = 1.0f;    // sec_mask (bool as 1.0f)
}

// ---------------------------------------------------------------------------
extern "C" void kernel_launch(void* const* d_in, const int* in_sizes, int n_in,
                              void* d_out, int out_size, void* d_ws, size_t ws_size,
                              hipStream_t stream) {
  const float* in = (const float*)d_in[0];
  const int S = in_sizes[0] / EXPERTS;  // 8192

  // capacity: floor(1.25 * S / E), made even, min 4  -> 160 for (8192, 64)
  int cap = (int)((double)S * 1.25 / (double)EXPERTS);
  cap += cap & 1;
  if (cap < 4) cap = 4;

  int*   top1 = (int*)d_ws;
  int*   slot = top1 + S;
  float* gate = (float*)(slot + S);

  float* out = (float*)d_out;
  size_t combineElems = (size_t)S * EXPERTS * (size_t)cap;
  long long n4 = (long long)(2 * combineElems) / 4;  // divisible: cap is even

  // K1: wave per token (8192 waves)
  {
    int threads = 256;
    int blocks  = (S * 32 + threads - 1) / threads;
    router_gate_kernel<<<blocks, threads, 0, stream>>>(in, top1, gate, S);
  }
  // K2: one block per expert, ordered scan with async LDS staging
  rank_scan_kernel<<<EXPERTS, 256, 0, stream>>>(top1, slot, S, cap);

  // K3: NT zero-fill of the full 671MB output
  zero_fill_kernel<<<4096, 256, 0, stream>>>(out, n4);

  // K4: sparse scatter of the nonzeros
  scatter_kernel<<<(S + 255) / 256, 256, 0, stream>>>(top1, gate, slot, out,
                                                      S, cap, combineElems);
}